// MambaWrapper_10849087389981
// MI455X (gfx1250) — compile-verified
//
#include <hip/hip_runtime.h>
#include <hip/hip_bf16.h>
#include <math.h>

// ---------------- problem constants ----------------
#define BATCH 4
#define LSEQ  4096
#define DM    512
#define DI    1024
#define DS    16
#define DC    4
#define DTR   32
#define MROWS (BATCH * LSEQ)     // 16384

// ---------------- WMMA types ----------------
typedef float v2f __attribute__((ext_vector_type(2)));
typedef float v8f __attribute__((ext_vector_type(8)));
typedef int   vi4 __attribute__((vector_size(16)));   // async-LDS builtin operand type

#define AS1 __attribute__((address_space(1)))
#define AS3 __attribute__((address_space(3)))

// Probe for the CDNA5 async global->LDS path (ASYNCcnt-tracked DMA).
#if defined(__has_builtin)
#if __has_builtin(__builtin_amdgcn_global_load_async_to_lds_b128) && \
    __has_builtin(__builtin_amdgcn_s_wait_asynccnt)
#define HAVE_ASYNC_LDS 1
#endif
#endif
#ifndef HAVE_ASYNC_LDS
#define HAVE_ASYNC_LDS 0
#endif

// ---------------- generic f32 WMMA GEMM ----------------
// C[M,N] = epilogue( A[M,K](row-major, stride lda) * W[N,K]^T + bias )
// Tiles: BM=64, BN=64, BK=16. 256 threads = 8 waves arranged 4x2,
// each wave owns two 16x16 output tiles (32 cols). LDS double-buffered
// when the async-to-LDS path is available.
#define BM 64
#define BN 64
#define BK 16
#define LDS_STR 20   // padded stride (floats): keeps 16B alignment, spreads banks

#define EPI_NONE      0
#define EPI_SOFTPLUS  1
#define EPI_GELU      2

__device__ __forceinline__ void wmma_compute_tile(const float* __restrict__ As,
                                                  const float* __restrict__ Bs,
                                                  int mo, int no, int lo, int hi,
                                                  v8f& acc0, v8f& acc1)
{
#pragma unroll
    for (int kk = 0; kk < BK; kk += 4) {
        // A fragment (16x4): lane lo = row, VGPR v holds k = kk + 2*hi + v
        v2f a = *(const v2f*)&As[(mo + lo) * LDS_STR + kk + 2 * hi];
        // B fragment (4x16): lane lo = col n, VGPR v holds k = kk + 2*v + hi
        v2f b0, b1;
        b0.x = Bs[(no + lo) * LDS_STR + kk + hi];
        b0.y = Bs[(no + lo) * LDS_STR + kk + 2 + hi];
        b1.x = Bs[(no + 16 + lo) * LDS_STR + kk + hi];
        b1.y = Bs[(no + 16 + lo) * LDS_STR + kk + 2 + hi];
        acc0 = __builtin_amdgcn_wmma_f32_16x16x4_f32(false, a, false, b0,
                                                     (short)0, acc0, false, false);
        acc1 = __builtin_amdgcn_wmma_f32_16x16x4_f32(false, a, false, b1,
                                                     (short)0, acc1, false, false);
    }
}

template <int EPI>
__global__ void __launch_bounds__(256)
gemm_wmma_f32(const float* __restrict__ A, int lda,
              const float* __restrict__ W,
              const float* __restrict__ bias,
              float* __restrict__ C,
              int M, int N, int K)
{
    __shared__ float As[2][BM * LDS_STR];
    __shared__ float Bs[2][BN * LDS_STR];

    const int tid  = threadIdx.x;
    const int lane = tid & 31;
    const int wave = tid >> 5;
    const int hi   = lane >> 4;
    const int lo   = lane & 15;

    const int mblk = blockIdx.y * BM;
    const int nblk = blockIdx.x * BN;
    const int wr = wave & 3;
    const int wc = wave >> 2;
    const int mo = wr * 16;
    const int no = wc * 32;

    v8f acc0 = {};
    v8f acc1 = {};

    // cooperative tile load mapping: thread t -> row t/4, 16B at col (t%4)*4
    const int lr = tid >> 2;
    const int lc = (tid & 3) * 4;
    const float* ga = A + (size_t)(mblk + lr) * lda + lc;   // + kt
    const float* gw = W + (size_t)(nblk + lr) * K   + lc;   // + kt
    const int lofs = lr * LDS_STR + lc;

#if HAVE_ASYNC_LDS
    // ---- async double-buffered pipeline: one barrier per K-tile ----
    __builtin_amdgcn_global_load_async_to_lds_b128(
        (AS1 vi4*)ga, (AS3 vi4*)&As[0][lofs], 0, 0);
    __builtin_amdgcn_global_load_async_to_lds_b128(
        (AS1 vi4*)gw, (AS3 vi4*)&Bs[0][lofs], 0, 0);
    __builtin_amdgcn_s_wait_asynccnt(0);
    __syncthreads();
    const int ntiles = K / BK;
    for (int t = 0; t < ntiles; ++t) {
        const int cur = t & 1;
        if (t + 1 < ntiles) {
            const int kt = (t + 1) * BK;
            __builtin_amdgcn_global_load_async_to_lds_b128(
                (AS1 vi4*)(ga + kt), (AS3 vi4*)&As[1 - cur][lofs], 0, 0);
            __builtin_amdgcn_global_load_async_to_lds_b128(
                (AS1 vi4*)(gw + kt), (AS3 vi4*)&Bs[1 - cur][lofs], 0, 0);
        }
        wmma_compute_tile(As[cur], Bs[cur], mo, no, lo, hi, acc0, acc1);
        __builtin_amdgcn_s_wait_asynccnt(0);
        __syncthreads();
    }
#else
    // ---- fallback: load to VGPRs, store to LDS ----
    for (int kt = 0; kt < K; kt += BK) {
        const float4 av = *(const float4*)(ga + kt);
        const float4 bv = *(const float4*)(gw + kt);
        *(float4*)&As[0][lofs] = av;
        *(float4*)&Bs[0][lofs] = bv;
        __syncthreads();
        wmma_compute_tile(As[0], Bs[0], mo, no, lo, hi, acc0, acc1);
        __syncthreads();
    }
#endif

    // C/D layout: VGPR v -> row = v + 8*hi, col = lo
#pragma unroll
    for (int v = 0; v < 8; ++v) {
        const int rm  = mblk + mo + v + 8 * hi;
        const int cn0 = nblk + no + lo;
        float x0 = acc0[v];
        float x1 = acc1[v];
        if (EPI == EPI_SOFTPLUS) {
            x0 += bias[cn0];
            x1 += bias[cn0 + 16];
            x0 = (x0 > 20.f) ? x0 : log1pf(expf(x0));
            x1 = (x1 > 20.f) ? x1 : log1pf(expf(x1));
        } else if (EPI == EPI_GELU) {
            x0 += bias[cn0];
            x1 += bias[cn0 + 16];
            x0 = 0.5f * x0 * (1.f + erff(x0 * 0.70710678118654752f));
            x1 = 0.5f * x1 * (1.f + erff(x1 * 0.70710678118654752f));
        }
        C[(size_t)rm * N + cn0]      = x0;
        C[(size_t)rm * N + cn0 + 16] = x1;
    }
}

// ---------------- transpose x [B, DM, L] -> xm [B*L, DM] ----------------
__global__ void __launch_bounds__(256)
transpose_x(const float* __restrict__ x, float* __restrict__ xm)
{
    __shared__ float tile[32][33];
    const int b  = blockIdx.z;
    const int l0 = blockIdx.x * 32;
    const int d0 = blockIdx.y * 32;
    const int tx = threadIdx.x, ty = threadIdx.y;
#pragma unroll
    for (int i = ty; i < 32; i += 8)
        tile[i][tx] = x[((size_t)b * DM + d0 + i) * LSEQ + l0 + tx];
    __syncthreads();
#pragma unroll
    for (int i = ty; i < 32; i += 8) {
        const int l = l0 + i, d = d0 + tx;
        xm[((size_t)b * LSEQ + l) * DM + d] = tile[tx][i];
    }
}

// ---------------- causal depthwise conv1d (k=4) + SiLU ----------------
__global__ void __launch_bounds__(256)
conv1d_silu(const float* __restrict__ xz, const float* __restrict__ w,
            const float* __restrict__ bias, float* __restrict__ xc)
{
    const int idx = blockIdx.x * 256 + threadIdx.x;    // over M*DI
    const int d = idx & (DI - 1);
    const int m = idx >> 10;
    const int l = m & (LSEQ - 1);
    float acc = bias[d];
#pragma unroll
    for (int k = 0; k < DC; ++k) {
        const int ls = l + k - (DC - 1);
        if (ls >= 0)
            acc += w[d * DC + k] * xz[(size_t)(m + k - (DC - 1)) * (2 * DI) + d];
    }
    acc = acc / (1.f + expf(-acc));                    // SiLU
    xc[idx] = acc;
}

// ---------------- selective scan ----------------
// one lane per (b, d, n) scalar recurrence; y reduced over n via shfl_xor(width 16)
__global__ void __launch_bounds__(256)
selective_scan(const float* __restrict__ dtv,    // [M, DI]
               const float* __restrict__ xc,     // [M, DI]
               const float* __restrict__ xdbl,   // [M, 64]  (dt|B|C)
               const float* __restrict__ A_log,  // [DI, DS]
               float* __restrict__ yv)           // [M, DI]
{
    const int g = blockIdx.x * 256 + threadIdx.x;      // 0 .. B*DI*DS-1
    const int n = g & 15;
    const int d = (g >> 4) & (DI - 1);
    const int b = g >> 14;
    const float A_dn = -expf(A_log[d * DS + n]);

    float h = 0.f;
    const size_t mbase = (size_t)b * LSEQ;
    for (int l = 0; l < LSEQ; ++l) {
        const size_t m = mbase + l;
        const float dt = dtv[m * DI + d];
        const float xv = xc[m * DI + d];
        const float Bn = xdbl[m * 64 + DTR + n];
        const float Cn = xdbl[m * 64 + DTR + DS + n];
        h = expf(dt * A_dn) * h + (dt * xv) * Bn;
        float part = h * Cn;
        part += __shfl_xor(part, 1, 16);
        part += __shfl_xor(part, 2, 16);
        part += __shfl_xor(part, 4, 16);
        part += __shfl_xor(part, 8, 16);
        if (n == 0) yv[m * DI + d] = part;
    }
}

// ---------------- y = (y + D*xc) * silu(z) ----------------
__global__ void __launch_bounds__(256)
gate_fuse(float* __restrict__ yv, const float* __restrict__ xc,
          const float* __restrict__ xz, const float* __restrict__ Dp)
{
    const int idx = blockIdx.x * 256 + threadIdx.x;
    const int d = idx & (DI - 1);
    const int m = idx >> 10;
    const float z = xz[(size_t)m * (2 * DI) + DI + d];
    const float y = yv[idx] + Dp[d] * xc[idx];
    yv[idx] = y * (z / (1.f + expf(-z)));
}

// ---------------- GroupNorm(1, DM) over (C, L) per sample ----------------
__global__ void stats_init(float* __restrict__ stats)
{
    if (threadIdx.x < 2 * BATCH) stats[threadIdx.x] = 0.f;
}

__global__ void __launch_bounds__(256)
gn_reduce(const float* __restrict__ g, float* __restrict__ stats)
{
    const int row0 = blockIdx.x * 8;
    const int b = row0 >> 12;                    // 4096 rows per batch
    float s = 0.f, ss = 0.f;
    for (int i = threadIdx.x; i < 8 * DM; i += 256) {
        const float v = g[(size_t)row0 * DM + i];
        s += v; ss += v * v;
    }
    __shared__ float r1[256], r2[256];
    r1[threadIdx.x] = s; r2[threadIdx.x] = ss;
    __syncthreads();
    for (int off = 128; off > 0; off >>= 1) {
        if (threadIdx.x < off) {
            r1[threadIdx.x] += r1[threadIdx.x + off];
            r2[threadIdx.x] += r2[threadIdx.x + off];
        }
        __syncthreads();
    }
    if (threadIdx.x == 0) {
        atomicAdd(&stats[b * 2 + 0], r1[0]);
        atomicAdd(&stats[b * 2 + 1], r2[0]);
    }
}

__global__ void __launch_bounds__(256)
gn_norm_transpose(const float* __restrict__ g, const float* __restrict__ stats,
                  const float* __restrict__ gamma, const float* __restrict__ beta,
                  float* __restrict__ out)
{
    __shared__ float tile[32][33];
    const int b  = blockIdx.z;
    const int o0 = blockIdx.x * 32;
    const int l0 = blockIdx.y * 32;
    const int tx = threadIdx.x, ty = threadIdx.y;
    const float cnt = (float)DM * (float)LSEQ;
    const float mu  = stats[b * 2] / cnt;
    const float var = stats[b * 2 + 1] / cnt - mu * mu;
    const float inv = rsqrtf(var + 1e-5f);
#pragma unroll
    for (int i = ty; i < 32; i += 8)
        tile[i][tx] = g[((size_t)b * LSEQ + l0 + i) * DM + o0 + tx];
    __syncthreads();
#pragma unroll
    for (int i = ty; i < 32; i += 8) {
        const int o = o0 + i, l = l0 + tx;
        const float v = (tile[tx][i] - mu) * inv * gamma[o] + beta[o];
        out[(size_t)b * DM * LSEQ + (size_t)o * LSEQ + l] = v;
    }
}

// ---------------- launch ----------------
extern "C" void kernel_launch(void* const* d_in, const int* in_sizes, int n_in,
                              void* d_out, int out_size, void* d_ws, size_t ws_size,
                              hipStream_t stream)
{
    const float* x         = (const float*)d_in[0];
    const float* in_proj_w = (const float*)d_in[1];
    const float* conv1d_w  = (const float*)d_in[2];
    const float* conv1d_b  = (const float*)d_in[3];
    const float* x_proj_w  = (const float*)d_in[4];
    const float* dt_proj_w = (const float*)d_in[5];
    const float* dt_proj_b = (const float*)d_in[6];
    const float* A_log     = (const float*)d_in[7];
    const float* Dp        = (const float*)d_in[8];
    const float* out_proj_w= (const float*)d_in[9];
    const float* conv_w    = (const float*)d_in[10];
    const float* conv_b    = (const float*)d_in[11];
    const float* gamma     = (const float*)d_in[12];
    const float* beta      = (const float*)d_in[13];
    float* out = (float*)d_out;

    float* ws = (float*)d_ws;
    const size_t MBf = 262144;
    float* xm   = ws;                  // [M, DM]    32 MB  (reused by out1)
    float* xz   = ws + 32  * MBf;      // [M, 2*DI] 128 MB  (first 32MB reused by g)
    float* xc   = ws + 160 * MBf;      // [M, DI]    64 MB
    float* dtv  = ws + 224 * MBf;      // [M, DI]    64 MB
    float* yv   = ws + 288 * MBf;      // [M, DI]    64 MB
    float* xdbl = ws + 352 * MBf;      // [M, 64]     4 MB
    float* stats= ws + 356 * MBf;      // 8 floats
    float* out1 = xm;
    float* gbuf = xz;

    dim3 blk256(256);

    transpose_x<<<dim3(LSEQ / 32, DM / 32, BATCH), dim3(32, 8), 0, stream>>>(x, xm);

    gemm_wmma_f32<EPI_NONE><<<dim3(2 * DI / BN, MROWS / BM), blk256, 0, stream>>>(
        xm, DM, in_proj_w, nullptr, xz, MROWS, 2 * DI, DM);

    conv1d_silu<<<dim3(MROWS * DI / 256), blk256, 0, stream>>>(xz, conv1d_w, conv1d_b, xc);

    gemm_wmma_f32<EPI_NONE><<<dim3(64 / BN, MROWS / BM), blk256, 0, stream>>>(
        xc, DI, x_proj_w, nullptr, xdbl, MROWS, 64, DI);

    gemm_wmma_f32<EPI_SOFTPLUS><<<dim3(DI / BN, MROWS / BM), blk256, 0, stream>>>(
        xdbl, 64, dt_proj_w, dt_proj_b, dtv, MROWS, DI, DTR);

    selective_scan<<<dim3(BATCH * DI * DS / 256), blk256, 0, stream>>>(
        dtv, xc, xdbl, A_log, yv);

    gate_fuse<<<dim3(MROWS * DI / 256), blk256, 0, stream>>>(yv, xc, xz, Dp);

    gemm_wmma_f32<EPI_NONE><<<dim3(DM / BN, MROWS / BM), blk256, 0, stream>>>(
        yv, DI, out_proj_w, nullptr, out1, MROWS, DM, DI);

    gemm_wmma_f32<EPI_GELU><<<dim3(DM / BN, MROWS / BM), blk256, 0, stream>>>(
        out1, DM, conv_w, conv_b, gbuf, MROWS, DM, DM);

    stats_init<<<1, 32, 0, stream>>>(stats);
    gn_reduce<<<dim3(MROWS / 8), blk256, 0, stream>>>(gbuf, stats);

    gn_norm_transpose<<<dim3(DM / 32, LSEQ / 32, BATCH), dim3(32, 8), 0, stream>>>(
        gbuf, stats, gamma, beta, out);
}